// MyLoss_20684562497962
// MI455X (gfx1250) — compile-verified
//
#include <hip/hip_runtime.h>
#include <math.h>

#define NBATCH 16
#define NANCH  3
#define NGRID  64
#define NT     60
#define GSZ    (NGRID*NGRID)        // 4096
#define NPB    (NANCH*GSZ)          // 12288 preds per batch
#define NLBL   (NBATCH*NT)          // 960 labels
#define BLOCKS_PER_B (NPB/256)      // 48

typedef float    v2f  __attribute__((ext_vector_type(2)));
typedef float    v8f  __attribute__((ext_vector_type(8)));
typedef _Float16 v16h __attribute__((ext_vector_type(16)));

struct WSData {
  float loss_xy_num, loss_wh_num, cnt, ig, pg;
  float sum_gt_gou, sum_iou_best, hm_sum;
  float pb_mg[NBATCH], pb_mi[NBATCH], pb_c2[NBATCH];
  float pb_inter[NBATCH], pb_pa[NBATCH], pb_ga[NBATCH];
  int   inds[NLBL];
  unsigned char obj[NBATCH*NPB];
  unsigned char maskp[NBATCH*NPB];
};

__device__ __forceinline__ float sigm(float x){ return 1.0f/(1.0f+expf(-x)); }

__device__ __forceinline__ void iou_gou(float px,float py,float pw,float ph,
                                        float gx,float gy,float gw,float gh,
                                        float& iou, float& gou){
  float px1=px-pw*0.5f, px2=px+pw*0.5f, py1=py-ph*0.5f, py2=py+ph*0.5f;
  float gx1=gx-gw*0.5f, gx2=gx+gw*0.5f, gy1=gy-gh*0.5f, gy2=gy+gh*0.5f;
  float iw=fmaxf(fminf(px2,gx2)-fmaxf(px1,gx1),0.0f);
  float ih=fmaxf(fminf(py2,gy2)-fmaxf(py1,gy1),0.0f);
  float inter=iw*ih;
  float pa=pw*ph, ga=gw*gh;
  float uni=pa+ga-inter;
  iou=inter/(uni+1e-16f);
  float cw=fmaxf(px2,gx2)-fminf(px1,gx1);
  float ch=fmaxf(py2,gy2)-fminf(py1,gy1);
  float ca=cw*ch;
  gou=iou-(ca-uni)/(ca+1e-16f);
}

__device__ __forceinline__ void pred_box(const float* out,int b,int n,
                                         float&px,float&py,float&pw,float&ph){
  int a=n>>12, rem=n&4095, j=rem>>6, i=rem&63;
  int base=((b*(NANCH*5)+a*5)*NGRID+j)*NGRID+i;
  px=(sigm(out[base])      +(float)i)*(1.0f/NGRID);
  py=(sigm(out[base+GSZ])  +(float)j)*(1.0f/NGRID);
  pw= sigm(out[base+2*GSZ]);
  ph= sigm(out[base+3*GSZ]);
}

__device__ __forceinline__ float bred(float v, float* s){
  int t=threadIdx.x; s[t]=v; __syncthreads();
  for(int k=128;k>0;k>>=1){ if(t<k) s[t]+=s[t+k]; __syncthreads(); }
  float r=s[0]; __syncthreads(); return r;
}

__global__ void k_zero(unsigned int* p,int words){
  int i=blockIdx.x*blockDim.x+threadIdx.x;
  if(i<words) p[i]=0u;
}

// build_target: per-label cell assignment with duplicate-key counting.
__global__ void k_build_target(const float* out,const float* labels,WSData* ws){
  int i=blockIdx.x*blockDim.x+threadIdx.x;
  if(i>=NLBL) return;
  const float* L=labels+i*5;
  int b=(int)L[0];
  float gx=L[1]*NGRID, gy=L[2]*NGRID;
  int gi=(int)gx, gj=(int)gy;
  int key=(b*NGRID+gj)*NGRID+gi;
  int occ=0;
  for(int jlb=0;jlb<i;jlb++){
    const float* Lj=labels+jlb*5;
    int kj=(((int)Lj[0])*NGRID+(int)(Lj[2]*NGRID))*NGRID+(int)(Lj[1]*NGRID);
    if(kj==key) occ++;
  }
  if(occ>=NANCH) return;           // a == nA plane is dropped by [:, :nA]
  int a=occ;
  float tx=gx-floorf(gx), ty=gy-floorf(gy);
  float tw=L[3], th=L[4];          // gw/nG == raw label w,h
  int base=((b*(NANCH*5)+a*5)*NGRID+gj)*NGRID+gi;
  float xs=sigm(out[base]);
  float ys=sigm(out[base+GSZ]);
  float wv=sigm(out[base+2*GSZ]);
  float hv=sigm(out[base+3*GSZ]);
  float cf=sigm(out[base+4*GSZ]);
  atomicAdd(&ws->loss_xy_num,(xs-tx)*(xs-tx)+(ys-ty)*(ys-ty));
  float dw=logf(wv)-logf(tw), dh=logf(hv)-logf(th);
  atomicAdd(&ws->loss_wh_num,dw*dw+dh*dh);
  atomicAdd(&ws->cnt,1.0f);
  atomicAdd(&ws->ig,cf);
  ws->obj[b*NPB + a*GSZ + gj*NGRID + gi]=1;
}

// One block per (b,t): argmax of GIoU over all 12288 preds (first-max ties).
__global__ void k_gt_reduce(const float* out,const float* labels,WSData* ws){
  __shared__ float sv[256]; __shared__ float sw[256]; __shared__ int sx[256];
  int bt=blockIdx.x; int b=bt/NT;
  const float* L=labels+bt*5;
  float gx=L[1],gy=L[2],gw=L[3],gh=L[4];
  float best=-1e30f, biou=0.f; int bidx=0x7fffffff;
  for(int n=threadIdx.x;n<NPB;n+=256){
    float px,py,pw,ph; pred_box(out,b,n,px,py,pw,ph);
    float iou,gou; iou_gou(px,py,pw,ph,gx,gy,gw,gh,iou,gou);
    if(gou>best){best=gou;biou=iou;bidx=n;}   // ascending n keeps first max
  }
  int t=threadIdx.x;
  sv[t]=best; sw[t]=biou; sx[t]=bidx;
  __syncthreads();
  for(int k=128;k>0;k>>=1){
    if(t<k){
      float ov=sv[t+k];
      if(ov>sv[t] || (ov==sv[t] && sx[t+k]<sx[t])){
        sv[t]=ov; sw[t]=sw[t+k]; sx[t]=sx[t+k];
      }
    }
    __syncthreads();
  }
  if(t==0){
    ws->inds[bt]=sx[0];
    atomicAdd(&ws->sum_gt_gou,1.0f-sv[0]);
    atomicAdd(&ws->sum_iou_best,1.0f-sw[0]);
  }
}

__global__ void k_scatter(WSData* ws){
  int i=blockIdx.x*blockDim.x+threadIdx.x;
  if(i<NLBL){ int b=i/NT; ws->maskp[b*NPB+ws->inds[i]]=1; }
}

// Per-pred: argmax over 60 GTs, then per-batch masked sums via block reductions.
__global__ void k_per_pred(const float* out,const float* labels,WSData* ws){
  __shared__ float sb[256];
  int b=blockIdx.x/BLOCKS_PER_B;
  int n=(blockIdx.x%BLOCKS_PER_B)*256+threadIdx.x;
  float px,py,pw,ph; pred_box(out,b,n,px,py,pw,ph);
  float best=-1e30f,biou=0.f;
  const float* Lb=labels+b*NT*5;
  for(int t=0;t<NT;t++){
    const float* L=Lb+t*5;
    float iou,gou; iou_gou(px,py,pw,ph,L[1],L[2],L[3],L[4],iou,gou);
    if(gou>best){best=gou;biou=iou;}          // first-max kept
  }
  int a=n>>12, rem=n&4095, j=rem>>6, i=rem&63;
  int base=((b*(NANCH*5)+a*5)*NGRID+j)*NGRID+i;
  float conf=sigm(out[base+4*GSZ]);
  bool mask = biou>=0.5f;
  bool mp   = ws->maskp[b*NPB+n]!=0;
  bool o    = ws->obj  [b*NPB+n]!=0;
  bool m2   = mask||mp;
  bool ign  = (!mask)||mp;
  float tconf=(o||mp)?1.0f:0.0f;
  float v;
  v=bred(m2?(1.0f-best):0.0f,sb); if(threadIdx.x==0) atomicAdd(&ws->pb_mg[b],v);
  v=bred(m2?(1.0f-biou):0.0f,sb); if(threadIdx.x==0) atomicAdd(&ws->pb_mi[b],v);
  v=bred(m2?1.0f:0.0f,sb);        if(threadIdx.x==0) atomicAdd(&ws->pb_c2[b],v);
  v=bred(ign?conf*tconf:0.0f,sb); if(threadIdx.x==0) atomicAdd(&ws->pb_inter[b],v);
  v=bred(ign?conf*conf:0.0f,sb);  if(threadIdx.x==0) atomicAdd(&ws->pb_pa[b],v);
  v=bred(ign?tconf:0.0f,sb);      if(threadIdx.x==0) atomicAdd(&ws->pb_ga[b],v);
  v=bred(conf*conf,sb);           if(threadIdx.x==0) atomicAdd(&ws->pg,v);
}

// Heatmap MSE reduction through the WMMA pipe.
// A = ones(16x4); B carries squared diffs. Sum of all 256 D entries == 16*sum(B)
// regardless of the lane->(k,n) mapping.
// Wave-uniform loop (readfirstlane), float2 (b64) loads, 4 independent
// accumulators -> 4 pipelined v_wmma per iteration, no per-lane guards.
__global__ void k_heatmap(const float* __restrict__ A, const float* __restrict__ B,
                          WSData* ws, int M){
  int lane = threadIdx.x & 31;
  int wave = __builtin_amdgcn_readfirstlane((int)((blockIdx.x*blockDim.x+threadIdx.x)>>5));
  int nw   = (int)((gridDim.x*blockDim.x)>>5);   // uniform (grid dims are scalar)
  const int PERW = 256;                          // elements per wave per step
  int step   = nw*PERW;                          // uniform
  int nsteps = M/step;                           // uniform (exact for M=786432, nw=512)

  v8f a0={0.f,0.f,0.f,0.f,0.f,0.f,0.f,0.f};
  v8f a1=a0, a2=a0, a3=a0;
  v2f ones; ones[0]=1.0f; ones[1]=1.0f;

  const v2f* __restrict__ A2=(const v2f*)A;
  const v2f* __restrict__ B2=(const v2f*)B;

  for(int s=0;s<nsteps;s++){
    int f0 = (wave*PERW + s*step)>>1;            // float2 index, uniform per wave
    v2f xa0=A2[f0+lane     ], xb0=B2[f0+lane     ];
    v2f xa1=A2[f0+lane+ 32 ], xb1=B2[f0+lane+ 32 ];
    v2f xa2=A2[f0+lane+ 64 ], xb2=B2[f0+lane+ 64 ];
    v2f xa3=A2[f0+lane+ 96 ], xb3=B2[f0+lane+ 96 ];
    v2f d0=xa0-xb0, d1=xa1-xb1, d2=xa2-xb2, d3=xa3-xb3;
    d0*=d0; d1*=d1; d2*=d2; d3*=d3;
    a0=__builtin_amdgcn_wmma_f32_16x16x4_f32(false,ones,false,d0,(short)0,a0,false,false);
    a1=__builtin_amdgcn_wmma_f32_16x16x4_f32(false,ones,false,d1,(short)0,a1,false,false);
    a2=__builtin_amdgcn_wmma_f32_16x16x4_f32(false,ones,false,d2,(short)0,a2,false,false);
    a3=__builtin_amdgcn_wmma_f32_16x16x4_f32(false,ones,false,d3,(short)0,a3,false,false);
  }
  v8f at=a0+a1+a2+a3;
  float s_=at[0]+at[1]+at[2]+at[3]+at[4]+at[5]+at[6]+at[7];

  // scalar tail (dead when step divides M exactly); pre-scale by 16 to match WMMA
  float extra=0.f;
  for(int i=nsteps*step + wave*32 + lane; i<M; i+=nw*32){
    float d=A[i]-B[i]; extra+=d*d;
  }
  s_ += 16.0f*extra;

  for(int off=16; off>0; off>>=1) s_ += __shfl_down(s_,(unsigned)off,32);
  if(lane==0) atomicAdd(&ws->hm_sum,s_);
}

__global__ void k_finalize(WSData* ws,float* out7){
  if(blockIdx.x==0 && threadIdx.x==0){
    float cnt=ws->cnt;
    float lxy=ws->loss_xy_num/cnt;
    float lwh=ws->loss_wh_num/cnt;
    float lgou=ws->sum_gt_gou/(float)NLBL;
    float liou=ws->sum_iou_best/(float)NLBL;
    float lcb=0.f, sg2=0.f, si2=0.f;
    for(int b=0;b<NBATCH;b++){
      sg2+=ws->pb_mg[b]/ws->pb_c2[b];
      si2+=ws->pb_mi[b]/ws->pb_c2[b];
      lcb+=1.0f-(2.0f*ws->pb_inter[b]+1.0f)/(ws->pb_pa[b]+ws->pb_ga[b]+1.0f);
    }
    lgou+=sg2/(float)NBATCH;
    liou+=si2/(float)NBATCH;
    lcb/=(float)NBATCH;
    float dice=(2.0f*ws->ig+1.0f)/(ws->pg+cnt+1.0f);
    if(dice!=dice) dice=1.0f;
    float lconf=1.0f-dice+lcb;
    float total=lxy+lwh+lconf+lgou;
    // WMMA over-counts by 16 (sum over the 16 unit rows of A)
    float hm=ws->hm_sum/(16.0f*(float)(NBATCH*NANCH*GSZ)*4.0f);
    out7[0]=lxy; out7[1]=lwh; out7[2]=lconf; out7[3]=liou;
    out7[4]=lgou; out7[5]=total; out7[6]=hm;
  }
}

extern "C" void kernel_launch(void* const* d_in, const int* in_sizes, int n_in,
                              void* d_out, int out_size, void* d_ws, size_t ws_size,
                              hipStream_t stream){
  (void)in_sizes; (void)n_in; (void)out_size; (void)ws_size;
  const float* out     =(const float*)d_in[0];
  const float* labels  =(const float*)d_in[1];
  const float* int_out =(const float*)d_in[2];
  const float* heatmaps=(const float*)d_in[3];
  WSData* ws=(WSData*)d_ws;

  int words=(int)((sizeof(WSData)+3u)/4u);
  k_zero        <<<(words+255)/256,256,0,stream>>>((unsigned int*)d_ws,words);
  k_build_target<<<(NLBL+255)/256,256,0,stream>>>(out,labels,ws);
  k_gt_reduce   <<<NLBL,256,0,stream>>>(out,labels,ws);
  k_scatter     <<<(NLBL+255)/256,256,0,stream>>>(ws);
  k_per_pred    <<<NBATCH*BLOCKS_PER_B,256,0,stream>>>(out,labels,ws);
  int M=4*NBATCH*NANCH*GSZ;   // 786432
  k_heatmap     <<<64,256,0,stream>>>(int_out,heatmaps,ws,M);
  k_finalize    <<<1,64,0,stream>>>(ws,(float*)d_out);
}